// GNN_v2_81664508166199
// MI455X (gfx1250) — compile-verified
//
#include <hip/hip_runtime.h>

#define NN 50000
#define NE 800000
#define EMB 128
#define HID 256
#define LAYERS 3
#define BN_EPS 1e-5f

typedef __attribute__((ext_vector_type(2))) float v2f;
typedef __attribute__((ext_vector_type(8))) float v8f;

// ---------------- zero fill ----------------
__global__ void gin_zero_kernel(float* __restrict__ p, int n) {
    int i = blockIdx.x * blockDim.x + threadIdx.x;
    if (i < n) p[i] = 0.0f;
}

// ---------------- input embedding: h = x_emb1[x0] + x_emb2[x1] ----------------
__global__ void gin_embed_kernel(const int* __restrict__ x,
                                 const float* __restrict__ e1,
                                 const float* __restrict__ e2,
                                 float* __restrict__ h) {
    int t = blockIdx.x * blockDim.x + threadIdx.x;     // NN*32 threads
    if (t >= NN * 32) return;
    int node = t >> 5;
    int q = (t & 31) * 4;
    int a0 = x[node * 2 + 0];
    int a1 = x[node * 2 + 1];
    float4 va = *(const float4*)(e1 + a0 * EMB + q);
    float4 vb = *(const float4*)(e2 + a1 * EMB + q);
    float4 o;
    o.x = va.x + vb.x; o.y = va.y + vb.y; o.z = va.z + vb.z; o.w = va.w + vb.w;
    *(float4*)(h + node * EMB + q) = o;
}

// ---------------- edge scatter: agg[dst] += h[src] + ee1[b0] + ee2[b1] ----------------
// one wave (32 lanes) per edge; each lane handles 4 contiguous floats (512B row total)
__global__ void gin_scatter_kernel(const int* __restrict__ ei,   // [2, NE]
                                   const int* __restrict__ ea,   // [NE, 2]
                                   const float* __restrict__ h,  // [NN, EMB]
                                   const float* __restrict__ ee1,// [6, EMB] (this layer)
                                   const float* __restrict__ ee2,// [3, EMB] (this layer)
                                   float* __restrict__ agg) {    // [NN, EMB]
    int w    = (blockIdx.x * blockDim.x + threadIdx.x) >> 5;
    int lane = threadIdx.x & 31;
    if (w >= NE) return;
    int src = ei[w];
    int dst = ei[NE + w];
    int b0  = ea[w * 2 + 0];
    int b1  = ea[w * 2 + 1];
    int q = lane * 4;
    float4 vh = *(const float4*)(h   + src * EMB + q);
    float4 v1 = *(const float4*)(ee1 + b0  * EMB + q);
    float4 v2 = *(const float4*)(ee2 + b1  * EMB + q);
    float* out = agg + dst * EMB + q;
    unsafeAtomicAdd(out + 0, vh.x + v1.x + v2.x);
    unsafeAtomicAdd(out + 1, vh.y + v1.y + v2.y);
    unsafeAtomicAdd(out + 2, vh.z + v1.z + v2.z);
    unsafeAtomicAdd(out + 3, vh.w + v1.w + v2.w);
}

// ---------------- fp32 WMMA GEMM: C[M,N] = relu?(A[M,K] @ B[K,N] + bias[N]) ----------------
// one wave per 16x16 output tile; V_WMMA_F32_16X16X4_F32 K-fragments
__global__ void gin_wmma_gemm_kernel(const float* __restrict__ A,
                                     const float* __restrict__ B,
                                     const float* __restrict__ bias,
                                     float* __restrict__ C,
                                     int M, int Nn, int K, int relu) {
    int wave = (blockIdx.x * blockDim.x + threadIdx.x) >> 5;
    int lane = threadIdx.x & 31;
    int ntiles = Nn >> 4;
    int tm = wave / ntiles;
    int tn = wave - tm * ntiles;
    if (tm * 16 >= M) return;                    // wave-uniform guard: EXEC stays all-ones

    int rowA = lane & 15;                        // A: M = lane%16
    int koff = (lane >> 4) * 2;                  // lanes 0-15: K0,K1 ; lanes 16-31: K2,K3
    const float* Arow = A + (size_t)(tm * 16 + rowA) * K;
    int col = tn * 16 + (lane & 15);             // B/D: N = lane%16

    v8f acc = {};
    #pragma unroll 4
    for (int k0 = 0; k0 < K; k0 += 4) {
        v2f a = *(const v2f*)(Arow + k0 + koff); // contiguous f32 pair, 8B aligned
        v2f b;
        b.x = B[(size_t)(k0 + koff + 0) * Nn + col];
        b.y = B[(size_t)(k0 + koff + 1) * Nn + col];
        acc = __builtin_amdgcn_wmma_f32_16x16x4_f32(
            false, a, false, b, (short)0, acc, false, false);
    }

    float bv = bias[col];
    int rbase = tm * 16 + ((lane >> 4) ? 8 : 0); // D: VGPR v -> M=v (lanes<16) / v+8
    #pragma unroll
    for (int v = 0; v < 8; ++v) {
        float o = acc[v] + bv;
        if (relu) o = fmaxf(o, 0.0f);
        C[(size_t)(rbase + v) * Nn + col] = o;
    }
}

// ---------------- BN stats: per-channel sum & sumsq with LDS reduction ----------------
__global__ void gin_bn_stats_kernel(const float* __restrict__ h,
                                    float* __restrict__ stats) { // [0:128]=sum [128:256]=sumsq
    __shared__ float s1[256];
    __shared__ float s2[256];
    int tid  = threadIdx.x;
    int c    = tid & 127;
    int half = tid >> 7;
    float acc = 0.0f, acc2 = 0.0f;
    for (int r = blockIdx.x * 2 + half; r < NN; r += gridDim.x * 2) {
        float v = h[(size_t)r * EMB + c];
        acc  += v;
        acc2 += v * v;
    }
    s1[tid] = acc;
    s2[tid] = acc2;
    __syncthreads();
    if (half == 0) {
        unsafeAtomicAdd(&stats[c],       s1[tid] + s1[tid + 128]);
        unsafeAtomicAdd(&stats[128 + c], s2[tid] + s2[tid + 128]);
    }
}

// ---------------- BN finalize: scale/shift per channel ----------------
__global__ void gin_bn_finalize_kernel(const float* __restrict__ stats,
                                       const float* __restrict__ gamma,
                                       const float* __restrict__ beta,
                                       float* __restrict__ ss) { // [0:128]=scale [128:256]=shift
    int c = threadIdx.x;
    if (c >= EMB) return;
    float invn = 1.0f / (float)NN;
    float mu   = stats[c] * invn;
    float var  = stats[128 + c] * invn - mu * mu;
    float sc   = gamma[c] * rsqrtf(var + BN_EPS);
    ss[c]       = sc;
    ss[128 + c] = beta[c] - mu * sc;
}

// ---------------- BN apply (+optional ReLU) ----------------
__global__ void gin_bn_apply_kernel(const float* __restrict__ in,
                                    float* __restrict__ out,
                                    const float* __restrict__ ss,
                                    int relu) {
    int t = blockIdx.x * blockDim.x + threadIdx.x;
    if (t >= NN * 32) return;
    int node = t >> 5;
    int q = (t & 31) * 4;
    float4 v = *(const float4*)(in + (size_t)node * EMB + q);
    float4 o;
    o.x = v.x * ss[q + 0] + ss[128 + q + 0];
    o.y = v.y * ss[q + 1] + ss[128 + q + 1];
    o.z = v.z * ss[q + 2] + ss[128 + q + 2];
    o.w = v.w * ss[q + 3] + ss[128 + q + 3];
    if (relu) {
        o.x = fmaxf(o.x, 0.0f); o.y = fmaxf(o.y, 0.0f);
        o.z = fmaxf(o.z, 0.0f); o.w = fmaxf(o.w, 0.0f);
    }
    *(float4*)(out + (size_t)node * EMB + q) = o;
}

extern "C" void kernel_launch(void* const* d_in, const int* in_sizes, int n_in,
                              void* d_out, int out_size, void* d_ws, size_t ws_size,
                              hipStream_t stream) {
    const int*   x         = (const int*)  d_in[0];
    const int*   edge_index= (const int*)  d_in[1];
    const int*   edge_attr = (const int*)  d_in[2];
    const float* x_emb1    = (const float*)d_in[3];
    const float* x_emb2    = (const float*)d_in[4];
    const float* edge_emb1 = (const float*)d_in[5];   // [3, 6, 128]
    const float* edge_emb2 = (const float*)d_in[6];   // [3, 3, 128]
    const float* W1        = (const float*)d_in[7];   // [3, 128, 256]
    const float* b1        = (const float*)d_in[8];   // [3, 256]
    const float* W2        = (const float*)d_in[9];   // [3, 256, 128]
    const float* b2        = (const float*)d_in[10];  // [3, 128]
    const float* gamma     = (const float*)d_in[11];  // [3, 128]
    const float* beta      = (const float*)d_in[12];  // [3, 128]
    float* out = (float*)d_out;

    // workspace carve-up
    char* ws = (char*)d_ws;
    float* hbuf  = (float*)(ws);                                        // NN*EMB
    float* agg   = (float*)(ws + (size_t)NN * EMB * 4);                 // NN*EMB
    float* tbuf  = (float*)(ws + (size_t)2 * NN * EMB * 4);             // NN*HID
    float* stats = (float*)(ws + (size_t)2 * NN * EMB * 4
                               + (size_t)NN * HID * 4);                 // 512 floats
    float* ss = stats + 256;                                            // scale/shift

    const int elem_threads = NN * 32;                   // 1.6M
    const int elem_blocks  = (elem_threads + 255) / 256;
    const int agg_elems    = NN * EMB;

    // input embedding -> hbuf
    gin_embed_kernel<<<elem_blocks, 256, 0, stream>>>(x, x_emb1, x_emb2, hbuf);

    for (int l = 0; l < LAYERS; ++l) {
        const float* ee1 = edge_emb1 + (size_t)l * 6 * EMB;
        const float* ee2 = edge_emb2 + (size_t)l * 3 * EMB;
        const float* w1l = W1 + (size_t)l * EMB * HID;
        const float* b1l = b1 + (size_t)l * HID;
        const float* w2l = W2 + (size_t)l * HID * EMB;
        const float* b2l = b2 + (size_t)l * EMB;
        const float* gl  = gamma + (size_t)l * EMB;
        const float* bl  = beta  + (size_t)l * EMB;

        // zero agg + stats
        gin_zero_kernel<<<(agg_elems + 255) / 256, 256, 0, stream>>>(agg, agg_elems);
        gin_zero_kernel<<<2, 256, 0, stream>>>(stats, 512);

        // scatter-add messages
        gin_scatter_kernel<<<(NE * 32 + 255) / 256, 256, 0, stream>>>(
            edge_index, edge_attr, hbuf, ee1, ee2, agg);

        // GEMM1: t = relu(agg @ W1 + b1)   [NN,128]x[128,256]
        {
            int waves  = (NN / 16) * (HID / 16);
            int blocks = (waves * 32 + 127) / 128;
            gin_wmma_gemm_kernel<<<blocks, 128, 0, stream>>>(
                agg, w1l, b1l, tbuf, NN, HID, EMB, 1);
        }
        // GEMM2: h = t @ W2 + b2           [NN,256]x[256,128]
        {
            int waves  = (NN / 16) * (EMB / 16);
            int blocks = (waves * 32 + 127) / 128;
            gin_wmma_gemm_kernel<<<blocks, 128, 0, stream>>>(
                tbuf, w2l, b2l, hbuf, NN, EMB, HID, 0);
        }

        // batchnorm
        gin_bn_stats_kernel<<<256, 256, 0, stream>>>(hbuf, stats);
        gin_bn_finalize_kernel<<<1, 128, 0, stream>>>(stats, gl, bl, ss);
        float* dst = (l == LAYERS - 1) ? out : hbuf;
        gin_bn_apply_kernel<<<elem_blocks, 256, 0, stream>>>(
            hbuf, dst, ss, (l != LAYERS - 1) ? 1 : 0);
    }
}